// JoinerGraphModel_68736656605510
// MI455X (gfx1250) — compile-verified
//
#include <hip/hip_runtime.h>
#include <math.h>
#include <stdint.h>

typedef __attribute__((ext_vector_type(2))) float v2f;
typedef __attribute__((ext_vector_type(8))) float v8f;

#define HDIM 128
#define WPB  8   // waves per block (256 threads)

// ---------------------------------------------------------------------------
// C[M x NC] = A[M x K] @ B[K x NC] + bias   (fp32, WMMA 16x16x4)
// B is staged in LDS K-pair-interleaved:  lds[(p*PSTR + n)*2 + (k&1)], p=k/2,
// so a WMMA B-fragment {B[k][n], B[k+1][n]} is ONE aligned ds_load_b64 whose
// destination pair feeds v_wmma directly (no register gluing).
// PSTR % 32 == 16  ->  the two lane-halves (K-offset 2*PSTR dwords == 32 banks)
// use disjoint bank sets; within a half, b64 loads sweep 32 consecutive banks.
// rows must be a multiple of 16 (50000 = 16*3125), K a multiple of 64.
// ---------------------------------------------------------------------------
template <int NC>
__global__ __launch_bounds__(256)
void gemm_bias_f32_wmma(const float* __restrict__ A,
                        const float* __restrict__ B,
                        const float* __restrict__ bias,
                        float* __restrict__ C,
                        int rowTiles, int K)
{
    constexpr int KC   = 64;           // K-chunk staged per iteration
    constexpr int PSTR = NC + 16;      // pair-row stride (floats): 144 / 80
    constexpr int COLT = NC / 16;      // column tiles
    constexpr int N4   = NC / 4;       // float4 per B row
    extern __shared__ float ldsB[];    // KC * PSTR floats

    const int tid  = threadIdx.x;
    const int wave = tid >> 5;
    const int lane = tid & 31;
    const int half = lane >> 4;        // 0: lanes 0-15, 1: lanes 16-31
    const int lm   = lane & 15;
    const int rowTile = blockIdx.x * WPB + wave;
    const bool active = rowTile < rowTiles;

    v8f acc[COLT];
    #pragma unroll
    for (int t = 0; t < COLT; ++t) { v8f z = {}; acc[t] = z; }

    // A-frag (f32 16x4): lane lm covers row rowTile*16+lm;
    // VGPR i holds A[m][k + 2*half + i] -> contiguous float2 (global b64).
    const float* arow = A + (size_t)((active ? rowTile : 0) * 16 + lm) * (size_t)K
                          + 2 * half;

    for (int k0 = 0; k0 < K; k0 += KC) {
        // ---- stage B chunk [KC x NC] into pair-interleaved LDS ----
        {
            const float4* Bg = (const float4*)(B + (size_t)k0 * NC);
            #pragma unroll
            for (int i = tid; i < (KC / 2) * N4; i += 256) {
                const int p  = i / N4;        // K-pair index
                const int n4 = i - p * N4;    // float4 column group
                const float4 g0 = Bg[(size_t)(2 * p)     * N4 + n4];
                const float4 g1 = Bg[(size_t)(2 * p + 1) * N4 + n4];
                float* dst = ldsB + ((size_t)p * PSTR + n4 * 4) * 2;
                v2f w;
                w.x = g0.x; w.y = g1.x; *(v2f*)(dst + 0) = w;  // ds_store_b64
                w.x = g0.y; w.y = g1.y; *(v2f*)(dst + 2) = w;
                w.x = g0.z; w.y = g1.z; *(v2f*)(dst + 4) = w;
                w.x = g0.w; w.y = g1.w; *(v2f*)(dst + 6) = w;
            }
        }
        // prefetch next A chunk while staging completes (global_prefetch)
        if (k0 + KC < K) __builtin_prefetch(arow + k0 + KC, 0, 0);
        __syncthreads();

        if (active) {
            #pragma unroll 2
            for (int kk = 0; kk < KC; kk += 4) {
                v2f a = *(const v2f*)(arow + k0 + kk);
                // fragment pair base: p = kk/2 + half, col = t*16 + lm
                const float* bp = ldsB + ((kk >> 1) + half) * (PSTR * 2) + lm * 2;
                #pragma unroll
                for (int t = 0; t < COLT; ++t) {
                    const v2f b = *(const v2f*)(bp + t * 32);  // one ds_load_b64
                    acc[t] = __builtin_amdgcn_wmma_f32_16x16x4_f32(
                        false, a, false, b, (short)0, acc[t], false, false);
                }
            }
        }
        __syncthreads();
    }

    if (active) {
        #pragma unroll
        for (int t = 0; t < COLT; ++t) {
            const int col = t * 16 + lm;
            const float bv = bias ? bias[col] : 0.0f;
            float* crow = C + (size_t)(rowTile * 16 + 8 * half) * NC + col;
            #pragma unroll
            for (int r = 0; r < 8; ++r) {
                // C/D layout: VGPR r -> row (r + 8*half), col = lane%16
                crow[(size_t)r * NC] = acc[t][r] + bv;
            }
        }
    }
}

// ---------------------------------------------------------------------------
// Edge phase: one wave per edge, 4 features per lane (float4 rows, 512B).
//   e    = ea * We[j] + be[j]                     (rank-1, computed on the fly)
//   gate = sigmoid(k[dst] + q[src] + 2e)
//   msg  = gate * (v[src] + e)
//   accum[dst] += msg    (accum pre-filled with h@Ws + b_gcn by GEMM)
// Node arrays are L2-resident (25.6 MB each, 192 MB L2).
// ---------------------------------------------------------------------------
__global__ __launch_bounds__(256)
void edge_msg_atomic(const float* __restrict__ kbuf,
                     const float* __restrict__ qbuf,
                     const float* __restrict__ vbuf,
                     const float* __restrict__ We_l,
                     const float* __restrict__ be_l,
                     const int*   __restrict__ ei,
                     const float* __restrict__ eattr,
                     float* __restrict__ accum,
                     int E)
{
    const int e = blockIdx.x * WPB + (threadIdx.x >> 5);
    if (e >= E) return;
    const int lane = threadIdx.x & 31;
    const int src  = ei[e];
    const int dst  = ei[E + e];
    const float ea = eattr[e];
    const int j0 = lane * 4;

    const float4 kd = *(const float4*)(kbuf + (size_t)dst * HDIM + j0);
    const float4 qs = *(const float4*)(qbuf + (size_t)src * HDIM + j0);
    const float4 vs = *(const float4*)(vbuf + (size_t)src * HDIM + j0);
    const float4 we = *(const float4*)(We_l + j0);
    const float4 be = *(const float4*)(be_l + j0);
    float* out = accum + (size_t)dst * HDIM + j0;

    const float* kp = (const float*)&kd;
    const float* qp = (const float*)&qs;
    const float* vp = (const float*)&vs;
    const float* wp = (const float*)&we;
    const float* bp = (const float*)&be;
    #pragma unroll
    for (int i = 0; i < 4; ++i) {
        const float ev = fmaf(ea, wp[i], bp[i]);
        float g = kp[i] + qp[i] + 2.0f * ev;
        g = 1.0f / (1.0f + __expf(-g));
        const float m = g * (vp[i] + ev);
        __hip_atomic_fetch_add(out + i, m, __ATOMIC_RELAXED,
                               __HIP_MEMORY_SCOPE_AGENT);
    }
}

// ---------------------------------------------------------------------------
// Node update: h = gelu(layernorm(accum) * gamma + beta).  One wave per node,
// 4 features per lane, cross-lane reductions via shfl_xor (wave32).
// ---------------------------------------------------------------------------
__device__ __forceinline__ float gelu_exact(float x)
{
    return 0.5f * x * (1.0f + erff(x * 0.70710678118654752f));
}

__global__ __launch_bounds__(256)
void node_ln_gelu(const float* __restrict__ sbuf,
                  const float* __restrict__ gamma_l,
                  const float* __restrict__ beta_l,
                  float* __restrict__ hout,
                  int N)
{
    const int node = blockIdx.x * WPB + (threadIdx.x >> 5);
    if (node >= N) return;
    const int lane = threadIdx.x & 31;
    const int j0 = lane * 4;

    const float4 xv = *(const float4*)(sbuf + (size_t)node * HDIM + j0);

    float s = xv.x + xv.y + xv.z + xv.w;
    #pragma unroll
    for (int off = 16; off > 0; off >>= 1) s += __shfl_xor(s, off, 32);
    const float mu = s * (1.0f / HDIM);

    const float d0 = xv.x - mu, d1 = xv.y - mu, d2 = xv.z - mu, d3 = xv.w - mu;
    float v = d0 * d0 + d1 * d1 + d2 * d2 + d3 * d3;
    #pragma unroll
    for (int off = 16; off > 0; off >>= 1) v += __shfl_xor(v, off, 32);
    const float rstd = rsqrtf(v * (1.0f / HDIM) + 1e-5f);

    const float4 g = *(const float4*)(gamma_l + j0);
    const float4 b = *(const float4*)(beta_l + j0);
    float4 o;
    o.x = gelu_exact(fmaf(d0 * rstd, g.x, b.x));
    o.y = gelu_exact(fmaf(d1 * rstd, g.y, b.y));
    o.z = gelu_exact(fmaf(d2 * rstd, g.z, b.z));
    o.w = gelu_exact(fmaf(d3 * rstd, g.w, b.w));
    *(float4*)(hout + (size_t)node * HDIM + j0) = o;
}

// ---------------------------------------------------------------------------
// Host orchestration (all launches on `stream`, graph-capture safe)
// ---------------------------------------------------------------------------
extern "C" void kernel_launch(void* const* d_in, const int* in_sizes, int n_in,
                              void* d_out, int out_size, void* d_ws, size_t ws_size,
                              hipStream_t stream)
{
    const float* x         = (const float*)d_in[0];
    const float* edge_attr = (const float*)d_in[1];
    const float* W_in      = (const float*)d_in[2];
    const float* b_in      = (const float*)d_in[3];
    const float* Wk        = (const float*)d_in[4];
    const float* bk        = (const float*)d_in[5];
    const float* Wq        = (const float*)d_in[6];
    const float* bq        = (const float*)d_in[7];
    const float* Wv        = (const float*)d_in[8];
    const float* bv        = (const float*)d_in[9];
    const float* We        = (const float*)d_in[10];
    const float* be        = (const float*)d_in[11];
    const float* Ws        = (const float*)d_in[12];
    const float* b_gcn     = (const float*)d_in[13];
    const float* gamma     = (const float*)d_in[14];
    const float* beta      = (const float*)d_in[15];
    const float* W_out     = (const float*)d_in[16];
    const int*   ei        = (const int*)d_in[17];

    const int D_IN = 256;
    const int N = in_sizes[0] / D_IN;     // 50000
    const int E = in_sizes[1];            // 800000
    const int L = 3;

    // workspace: h, k, q, v, s  (each N*H floats, ~25.6 MB)
    float* hbuf = (float*)d_ws;
    float* kbuf = hbuf + (size_t)N * HDIM;
    float* qbuf = kbuf + (size_t)N * HDIM;
    float* vbuf = qbuf + (size_t)N * HDIM;
    float* sbuf = vbuf + (size_t)N * HDIM;

    const int rowTiles = N / 16;                        // 3125 (exact)
    const int gemmGrid = (rowTiles + WPB - 1) / WPB;    // 391
    const size_t lds128 = 64 * (128 + 16) * sizeof(float);  // 36 KB
    const size_t lds64  = 64 * (64 + 16) * sizeof(float);   // 20 KB
    const int edgeGrid = (E + WPB - 1) / WPB;
    const int nodeGrid = (N + WPB - 1) / WPB;

    // h = x @ W_in + b_in
    gemm_bias_f32_wmma<128><<<gemmGrid, 256, lds128, stream>>>(
        x, W_in, b_in, hbuf, rowTiles, D_IN);

    for (int l = 0; l < L; ++l) {
        const size_t wo = (size_t)l * HDIM * HDIM;
        const size_t bo = (size_t)l * HDIM;
        gemm_bias_f32_wmma<128><<<gemmGrid, 256, lds128, stream>>>(
            hbuf, Wk + wo, bk + bo, kbuf, rowTiles, HDIM);
        gemm_bias_f32_wmma<128><<<gemmGrid, 256, lds128, stream>>>(
            hbuf, Wq + wo, bq + bo, qbuf, rowTiles, HDIM);
        gemm_bias_f32_wmma<128><<<gemmGrid, 256, lds128, stream>>>(
            hbuf, Wv + wo, bv + bo, vbuf, rowTiles, HDIM);
        // s = h @ Ws + b_gcn  (edge messages accumulate on top of this)
        gemm_bias_f32_wmma<128><<<gemmGrid, 256, lds128, stream>>>(
            hbuf, Ws + wo, b_gcn + bo, sbuf, rowTiles, HDIM);
        edge_msg_atomic<<<edgeGrid, 256, 0, stream>>>(
            kbuf, qbuf, vbuf, We + bo, be + bo, ei, edge_attr, sbuf, E);
        node_ln_gelu<<<nodeGrid, 256, 0, stream>>>(
            sbuf, gamma + bo, beta + bo, hbuf, N);
    }

    // out = h @ W_out  (no bias)
    gemm_bias_f32_wmma<64><<<gemmGrid, 256, lds64, stream>>>(
        hbuf, W_out, nullptr, (float*)d_out, rowTiles, HDIM);
}